// SO3PredictionModelGated_86088324481894
// MI455X (gfx1250) — compile-verified
//
#include <hip/hip_runtime.h>
#include <hip/hip_bf16.h>
#include <math.h>

// ---------------------------------------------------------------------------
// SO3 gated recurrent model, MI455X (gfx1250, wave32, WMMA).
//
// Key optimization: there is no nonlinearity between hid = quat@W_in^T and the
// five linear maps of hid, so everything pre-scan collapses to
//     out[t,c] = quat[t,:] @ (W_all @ W_in)[c,:]      (K = 4)
// Wc = W_all @ W_in is a 2432x4 matrix computed once with V_WMMA_F32_16X16X4_F32
// (fp32 WMMA, exact). The token projection is then one WMMA per 16x16 tile.
// The scan decomposes into B*NB = 1024 independent 4-wide quaternion
// recurrences -> 4 lanes per recurrence, rotor applied with __shfl_xor
// (Hamilton permutation is lane^1/lane^2/lane^3), fp16 streams double-buffered
// in registers in 8-step chunks + global_prefetch two chunks ahead.
// ---------------------------------------------------------------------------

#define Bq   8
#define Sq   8192
#define Dq   512
#define NBq  128
#define Tq   (Bq * Sq)          /* 65536 tokens            */
#define NCq  2432               /* 384 + 4*512 columns     */
#define CT_TILES 152            /* 2432 / 16               */

typedef float v8f __attribute__((ext_vector_type(8)));
typedef float v2f __attribute__((ext_vector_type(2)));
typedef _Float16 f16x8 __attribute__((ext_vector_type(8)));

// Column order in the fused weight matrix:
//   [0,384)      biv   (no activation)
//   [384,896)    gate  (sigmoid)
//   [896,1408)   rdec  (sigmoid)
//   [1408,1920)  ddec  (sigmoid)
//   [1920,2432)  inj   (no activation)
__device__ __forceinline__ const float* wall_row(int c,
    const float* Wb, const float* Wg, const float* Wr,
    const float* Wd, const float* Wi) {
  if (c < 384)  return Wb + (size_t)c * Dq;
  if (c < 896)  return Wg + (size_t)(c - 384) * Dq;
  if (c < 1408) return Wr + (size_t)(c - 896) * Dq;
  if (c < 1920) return Wd + (size_t)(c - 1408) * Dq;
  return Wi + (size_t)(c - 1920) * Dq;
}

// ---------------------------------------------------------------------------
// Kernel 1: Wc[2432][4] = W_all(2432x512) x W_in(512x4), fp32 WMMA 16x16x4.
// 152 row tiles, one wave each (19 blocks x 8 waves). K loop = 128 WMMAs.
// A frag (16x4 f32): lane<16 holds K={k,k+1}, lane>=16 holds K={k+2,k+3}.
// B frag (4x16 f32): lane n holds B[k'][n] for the same K split; cols >=4 are
// zeroed with a cndmask select (loads are unconditional + in-bounds so the
// compiler does not generate exec-masked branches around them).
// ---------------------------------------------------------------------------
__global__ void k1_combine_weights(const float* __restrict__ W_in,
                                   const float* __restrict__ W_biv,
                                   const float* __restrict__ W_gate,
                                   const float* __restrict__ W_rdec,
                                   const float* __restrict__ W_ddec,
                                   const float* __restrict__ W_inj,
                                   float* __restrict__ Wc) {
  const int wave = (blockIdx.x * blockDim.x + threadIdx.x) >> 5;   // 0..151
  const int lane = threadIdx.x & 31;
  const int hi   = lane >> 4;                                       // half-wave
  const int l    = lane & 15;
  const int q    = l & 3;                                           // wrapped col
  const float colmask = (l < 4) ? 1.0f : 0.0f;                      // cndmask
  const int row  = wave * 16 + l;                                   // < 2432
  const float* arow = wall_row(row, W_biv, W_gate, W_rdec, W_ddec, W_inj);

  v8f acc = {};
  for (int k = 0; k < Dq; k += 4) {
    const int k0 = k + hi * 2;
    v2f a; a.x = arow[k0]; a.y = arow[k0 + 1];
    // unconditional, in-bounds loads; zero columns >= 4 via select
    v2f bb;
    bb.x = W_in[(size_t)k0 * 4 + q] * colmask;
    bb.y = W_in[(size_t)(k0 + 1) * 4 + q] * colmask;
    acc = __builtin_amdgcn_wmma_f32_16x16x4_f32(
        false, a, false, bb, (short)0, acc, false, false);
  }
  // C layout: lane l<16 -> col l rows 0..7; lane 16+l -> col l rows 8..15.
  if (l < 4) {
    const int m0 = hi * 8;
    #pragma unroll
    for (int i = 0; i < 8; ++i)
      Wc[(size_t)(wave * 16 + m0 + i) * 4 + l] = acc[i];
  }
}

// ---------------------------------------------------------------------------
// Kernel 2: proj[c][t] (fp16) = sigma?( quat(Tx4) x Wc^T(4x2432) ).
// One WMMA (K=4) per 16x16 tile; each wave keeps one B frag and sweeps 8
// token tiles. Epilogue applies sigmoid to the gate/rdec/ddec column ranges
// and stores fp16 column-major over tokens (scan lanes stream contiguously).
// ---------------------------------------------------------------------------
__global__ void k2_project(const float* __restrict__ quat,
                           const float* __restrict__ Wc,
                           _Float16* __restrict__ proj) {
  const int wid  = (blockIdx.x * blockDim.x + threadIdx.x) >> 5;  // wave task
  const int lane = threadIdx.x & 31;
  const int hi   = lane >> 4;
  const int l    = lane & 15;

  const int ct = wid % CT_TILES;          // column tile
  const int tg = wid / CT_TILES;          // token-tile group of 8
  const int c0 = ct * 16;
  const int c  = c0 + l;                  // this lane's output column
  const bool sig = (c >= 384) && (c < 1920);

  // B frag: lane l holds Wc[c][k'] for k' = hi*2, hi*2+1
  v2f b;
  b.x = Wc[(size_t)c * 4 + hi * 2 + 0];
  b.y = Wc[(size_t)c * 4 + hi * 2 + 1];

  #pragma unroll
  for (int i = 0; i < 8; ++i) {
    const int t0 = (tg * 8 + i) * 16;
    // A frag: lane l (half hi) holds quat[t0+l][hi*2 .. hi*2+1]
    const float* qp = quat + (size_t)(t0 + l) * 4 + hi * 2;
    v2f a; a.x = qp[0]; a.y = qp[1];
    v8f acc = {};
    acc = __builtin_amdgcn_wmma_f32_16x16x4_f32(
        false, a, false, b, (short)0, acc, false, false);
    // lane holds col c, token rows t0 + hi*8 + (0..7)
    f16x8 o;
    #pragma unroll
    for (int m = 0; m < 8; ++m) {
      float v = acc[m];
      const float sv = 1.0f / (1.0f + __expf(-v));
      v = sig ? sv : v;                    // select, no divergence
      o[m] = (_Float16)v;
    }
    *(f16x8*)(proj + (size_t)c * Tq + t0 + hi * 8) = o;
  }
}

// ---------------------------------------------------------------------------
// Kernel 3: the serial scan. 1024 independent 4-wide quaternion recurrences.
// grid = (4, B), block = 128: thread -> (b, d), d = 4n + j. Lane groups of 4
// share the rotor/state via shuffles; Hamilton-product permutation for
// component j is exactly (j^1, j^2, j^3) -> __shfl_xor.
// Streams (gate/rdec/ddec/inj/biv) are fp16, column-contiguous over tokens,
// consumed in 8-step chunks with register double buffering and
// global_prefetch_b8 two chunks ahead to cover HBM/L2 latency.
// ---------------------------------------------------------------------------
__global__ void k3_scan(const _Float16* __restrict__ proj,
                        _Float16* __restrict__ hidden) {
  const int d = blockIdx.x * blockDim.x + threadIdx.x;   // 0..511
  const int b = blockIdx.y;
  const int n = d >> 2;
  const int j = d & 3;
  const int lane = threadIdx.x & 31;
  const int base = lane & ~3;

  // Hamilton signs for component j: hr_j = r0*h_j + sa*r1*h_{j^1}
  //                                        + sb*r2*h_{j^2} + sc*r3*h_{j^3}
  const float sa = (j & 1) ? 1.0f : -1.0f;
  const float sb = ((j ^ (j >> 1)) & 1) ? 1.0f : -1.0f;
  const float sc = (j >> 1) ? 1.0f : -1.0f;

  const size_t tb = (size_t)b * Sq;                      // token base
  const f16x8* gs = (const f16x8*)(proj + (size_t)(384 + d)  * Tq + tb);
  const f16x8* rs = (const f16x8*)(proj + (size_t)(896 + d)  * Tq + tb);
  const f16x8* ds = (const f16x8*)(proj + (size_t)(1408 + d) * Tq + tb);
  const f16x8* is = (const f16x8*)(proj + (size_t)(1920 + d) * Tq + tb);
  const int bc = 3 * n + ((j < 3) ? j : 2);              // j==3 stream unused
  const f16x8* bs = (const f16x8*)(proj + (size_t)bc * Tq + tb);

  const int CHUNKS = Sq / 8;
  f16x8 gC = gs[0], rC = rs[0], dC = ds[0], iC = is[0], bC = bs[0];
  float h = 0.0f;

  for (int ch = 0; ch < CHUNKS; ++ch) {
    f16x8 gN = {}, rN = {}, dN = {}, iN = {}, bN = {};
    if (ch + 1 < CHUNKS) {
      gN = gs[ch + 1]; rN = rs[ch + 1]; dN = ds[ch + 1];
      iN = is[ch + 1]; bN = bs[ch + 1];
    }
    if (ch + 2 < CHUNKS) {                 // global_prefetch_b8, 2 chunks ahead
      __builtin_prefetch(gs + ch + 2, 0, 1);
      __builtin_prefetch(rs + ch + 2, 0, 1);
      __builtin_prefetch(ds + ch + 2, 0, 1);
      __builtin_prefetch(is + ch + 2, 0, 1);
      __builtin_prefetch(bs + ch + 2, 0, 1);
    }
    #pragma unroll
    for (int k = 0; k < 8; ++k) {
      // bivector -> rotor (exp map); every lane reconstructs all 3 components
      const float bvj = (float)bC[k];
      const float b0 = __shfl(bvj, base + 0);
      const float b1 = __shfl(bvj, base + 1);
      const float b2 = __shfl(bvj, base + 2);
      const float ang = sqrtf(b0 * b0 + b1 * b1 + b2 * b2);
      const float w   = __cosf(ang);
      const float snc = (ang > 1e-8f) ? (__sinf(ang) / ang) : 1.0f;
      const float r1 = snc * b0, r2 = snc * b1, r3 = snc * b2;

      // gather the other three state components of this quaternion block
      const float ha = __shfl_xor(h, 1);
      const float hb = __shfl_xor(h, 2);
      const float hc = __shfl_xor(h, 3);
      const float hr = w * h + sa * r1 * ha + sb * r2 * hb + sc * r3 * hc;

      const float g  = (float)gC[k];
      const float rd = (float)rC[k];
      const float dd = (float)dC[k];
      const float ij = (float)iC[k];
      h = g * (rd * hr) + (1.0f - g) * (dd * h) + ij;

      hidden[(tb + (size_t)ch * 8 + k) * Dq + d] = (_Float16)h;
    }
    gC = gN; rC = rN; dC = dN; iC = iN; bC = bN;
  }
}

// ---------------------------------------------------------------------------
// Kernel 4: raw[t] = hidden[t,:] @ W_out^T (D->4), normalize. One wave/token.
// ---------------------------------------------------------------------------
__global__ void k4_output(const _Float16* __restrict__ hidden,
                          const float* __restrict__ W_out,
                          float* __restrict__ out) {
  const int t    = (blockIdx.x * blockDim.x + threadIdx.x) >> 5;  // token
  const int lane = threadIdx.x & 31;
  const int d0   = lane * 16;

  const f16x8* hp = (const f16x8*)(hidden + (size_t)t * Dq + d0);
  const f16x8 hA = hp[0], hB = hp[1];

  float acc[4];
  #pragma unroll
  for (int q = 0; q < 4; ++q) {
    const float* w = W_out + (size_t)q * Dq + d0;
    float s = 0.0f;
    #pragma unroll
    for (int i = 0; i < 8; ++i) s = fmaf((float)hA[i], w[i], s);
    #pragma unroll
    for (int i = 0; i < 8; ++i) s = fmaf((float)hB[i], w[8 + i], s);
    acc[q] = s;
  }
  #pragma unroll
  for (int off = 16; off >= 1; off >>= 1) {
    #pragma unroll
    for (int q = 0; q < 4; ++q) acc[q] += __shfl_xor(acc[q], off);
  }
  if (lane == 0) {
    const float nrm = sqrtf(acc[0] * acc[0] + acc[1] * acc[1] +
                            acc[2] * acc[2] + acc[3] * acc[3]);
    const float inv = 1.0f / fmaxf(nrm, 1e-8f);
    #pragma unroll
    for (int q = 0; q < 4; ++q) out[(size_t)t * 4 + q] = acc[q] * inv;
  }
}

// ---------------------------------------------------------------------------
// Workspace map (bytes):
//   [0, 38912)                        Wc (2432 x 4 fp32)
//   [65536, 65536 + 318767104)        proj (2432 x 65536 fp16)
//   [+318767104, +385875968)          hidden (65536 x 512 fp16)
// Total ~386 MB.
// ---------------------------------------------------------------------------
extern "C" void kernel_launch(void* const* d_in, const int* in_sizes, int n_in,
                              void* d_out, int out_size, void* d_ws, size_t ws_size,
                              hipStream_t stream) {
  const float* quat   = (const float*)d_in[0];
  const float* W_in   = (const float*)d_in[1];
  const float* W_biv  = (const float*)d_in[2];
  const float* W_gate = (const float*)d_in[3];
  const float* W_rdec = (const float*)d_in[4];
  const float* W_ddec = (const float*)d_in[5];
  const float* W_inj  = (const float*)d_in[6];
  const float* W_out  = (const float*)d_in[7];

  float*    Wc     = (float*)d_ws;
  _Float16* proj   = (_Float16*)((char*)d_ws + 65536);
  _Float16* hidden = (_Float16*)((char*)d_ws + 65536 + (size_t)NCq * Tq * 2);

  // 152 waves -> 19 blocks of 8 waves
  k1_combine_weights<<<19, 256, 0, stream>>>(W_in, W_biv, W_gate, W_rdec,
                                             W_ddec, W_inj, Wc);
  // (65536/16/8 token groups) * 152 col tiles = 77824 waves -> 9728 blocks
  k2_project<<<9728, 256, 0, stream>>>(quat, Wc, proj);
  // 8 batches x 512 chains-of-4 -> 32 blocks of 128 (4 waves/WGP)
  k3_scan<<<dim3(4, Bq), 128, 0, stream>>>(proj, hidden);
  // one wave per token
  k4_output<<<Tq / 8, 256, 0, stream>>>(hidden, W_out, (float*)d_out);
}